// OverlappingEdgeProcessor_82051055222863
// MI455X (gfx1250) — compile-verified
//
#include <hip/hip_runtime.h>

// (4, 6, 16, 512, 512) f32 cubemap edge blend for MI455X (gfx1250, wave32).
// Phase 1: streaming b128 nontemporal copy (memory-roofline bound).
// Phase 2: WMMA-based edge fixup: blend = ER*diag(w) + AR*diag(1-w) via two
//          chained V_WMMA_F32_16X16X4_F32 ops per 16-element tile.

typedef __attribute__((ext_vector_type(4))) float f4;
typedef __attribute__((ext_vector_type(2))) float v2f;
typedef __attribute__((ext_vector_type(8))) float v8f;

#if __has_builtin(__builtin_amdgcn_wmma_f32_16x16x4_f32)
#define HAS_WMMA_F32X4 1
#else
#define HAS_WMMA_F32X4 0
#endif

// ---------------- Phase 1: bulk streaming copy ----------------
__global__ __launch_bounds__(256) void bulk_copy_kernel(const f4* __restrict__ src,
                                                        f4* __restrict__ dst,
                                                        int n4) {
  const int stride = (int)(gridDim.x * blockDim.x);
  for (int i = (int)(blockIdx.x * blockDim.x + threadIdx.x); i < n4; i += stride) {
    f4 v = __builtin_nontemporal_load(src + i);
    __builtin_nontemporal_store(v, dst + i);
  }
}

// ---------------- Phase 2: edge fixup ----------------
// The 16 blends that survive the reference's shape check (ADJ entries whose
// edge axes match). Entries 0..7 are column strips (e=0/1), 8..15 row strips
// (e=2/3).  o0 = strip start in face, ao0 = strip start in adjacent face,
// wf = weight flipped (w10 instead of w01), mask = skip rows <4 / >=508
// (row edges of the same face overwrite those corners later in ref order).
__constant__ int kEf[16]    = {0, 0, 1, 1, 2, 2, 3, 3,  0, 0, 2, 2, 4, 4, 5, 5};
__constant__ int kEaf[16]   = {2, 3, 3, 2, 1, 0, 0, 1,  4, 5, 4, 5, 0, 2, 0, 2};
__constant__ int kEo0[16]   = {0, 508, 0, 508, 0, 508, 0, 508,  0, 508, 0, 508, 0, 508, 0, 508};
__constant__ int kEao0[16]  = {508, 0, 508, 0, 508, 0, 508, 0,  0, 0, 508, 508, 0, 0, 508, 508};
__constant__ int kEwf[16]   = {0, 1, 0, 1, 0, 1, 0, 1,  0, 1, 0, 1, 0, 1, 0, 1};
__constant__ int kEmask[16] = {1, 1, 0, 0, 1, 1, 0, 0,  0, 0, 0, 0, 0, 0, 0, 0};

__global__ __launch_bounds__(256) void edge_fixup_kernel(const float* __restrict__ x,
                                                         float* __restrict__ out) {
  // One wave32 per (edge, b, c, 16-wide tile along the 512 axis).
  // 16 edges * 4 B * 16 C * 32 tiles = 32768 waves.
  const int lane = (int)(threadIdx.x & 31);
  const int wave = (int)(blockIdx.x * (blockDim.x >> 5) + (threadIdx.x >> 5));

  const int tile = wave & 31;
  const int c    = (wave >> 5) & 15;
  const int b    = (wave >> 9) & 3;
  const int e    = (wave >> 11) & 15;

  const int f      = kEf[e];
  const int af     = kEaf[e];
  const int o0     = kEo0[e];
  const int ao0    = kEao0[e];
  const int wflip  = kEwf[e];
  const int domask = kEmask[e];
  const bool isRow = (e >= 8);

  const int half = lane >> 4;   // lane group (ISA VGPR striping)
  const int lid  = lane & 15;
  const int L0   = tile << 4;   // position along the 512-long axis

  const float* Pf = x + ((size_t)((b * 6 + f) * 16 + c) << 18);
  const float* Pa = x + ((size_t)((b * 6 + af) * 16 + c) << 18);
  float*       Qf = out + ((size_t)((b * 6 + f) * 16 + c) << 18);

  const float W01[4] = {0.0f, 1.0f / 3.0f, 2.0f / 3.0f, 1.0f};
  const int t0 = 2 * half, t1 = t0 + 1;
  const float w0 = wflip ? W01[3 - t0] : W01[t0];
  const float w1 = wflip ? W01[3 - t1] : W01[t1];
  const float u0 = 1.0f - w0, u1 = 1.0f - w1;

#if HAS_WMMA_F32X4
  v8f acc = {0.f, 0.f, 0.f, 0.f, 0.f, 0.f, 0.f, 0.f};
  v2f A, Bm;
  if (!isRow) {
    // Column strip: D(16x16) = AR(16x4)*diag(1-w) + ER(16x4)*diag(w).
    // A 16x4 layout: lane lid (+half*16): VGPR0=A[lid][t0], VGPR1=A[lid][t1].
    const float* pa = Pa + (L0 + lid) * 512 + ao0 + t0;
    A.x = pa[0];
    A.y = pa[1];
    // B 4x16 diag: VGPR0=B[t0][lid], VGPR1=B[t1][lid].
    Bm.x = (lid == t0) ? u0 : 0.0f;
    Bm.y = (lid == t1) ? u1 : 0.0f;
    acc = __builtin_amdgcn_wmma_f32_16x16x4_f32(false, A, false, Bm, (short)0, acc,
                                                false, false);
    const float* pe = Pf + (L0 + lid) * 512 + o0 + t0;
    A.x = pe[0];
    A.y = pe[1];
    Bm.x = (lid == t0) ? w0 : 0.0f;
    Bm.y = (lid == t1) ? w1 : 0.0f;
    acc = __builtin_amdgcn_wmma_f32_16x16x4_f32(false, A, false, Bm, (short)0, acc,
                                                false, false);
    // Valid output: columns N=0..3 -> lanes with lid<4 (both halves).
    // D VGPR r on this lane = D[r + 8*half][lid].
    if (lid < 4) {
      const int col = o0 + lid;
      #pragma unroll
      for (int r = 0; r < 8; ++r) {
        const int row = L0 + r + 8 * half;
        if (!domask || (row >= 4 && row < 508)) Qf[row * 512 + col] = acc[r];
      }
    }
  } else {
    // Row strip: D(16x16) = diag(1-w)*AR(4x16) + diag(w)*ER(4x16).
    // B 4x16 layout: lane lid (+half*16): VGPR0=B[t0][lid], VGPR1=B[t1][lid].
    const float* pa = Pa + (ao0 + t0) * 512 + L0 + lid;
    Bm.x = pa[0];
    Bm.y = pa[512];
    // A 16x4 diag: VGPR0=A[lid][t0], VGPR1=A[lid][t1].
    A.x = (lid == t0) ? u0 : 0.0f;
    A.y = (lid == t1) ? u1 : 0.0f;
    acc = __builtin_amdgcn_wmma_f32_16x16x4_f32(false, A, false, Bm, (short)0, acc,
                                                false, false);
    const float* pe = Pf + (o0 + t0) * 512 + L0 + lid;
    Bm.x = pe[0];
    Bm.y = pe[512];
    A.x = (lid == t0) ? w0 : 0.0f;
    A.y = (lid == t1) ? w1 : 0.0f;
    acc = __builtin_amdgcn_wmma_f32_16x16x4_f32(false, A, false, Bm, (short)0, acc,
                                                false, false);
    // Valid output: rows M=0..3 -> VGPRs 0..3 on lanes 0..15 (half==0).
    if (half == 0) {
      const int col = L0 + lid;
      #pragma unroll
      for (int r = 0; r < 4; ++r) Qf[(o0 + r) * 512 + col] = acc[r];
    }
  }
#else
  // Scalar fallback (identical math) so the build stays green if the builtin
  // signature probe fails.
  for (int q = lane; q < 64; q += 32) {
    const int t = q >> 4;   // offset within 4-wide strip
    const int n = q & 15;   // offset along the long axis (within tile)
    const float w = wflip ? W01[3 - t] : W01[t];
    if (!isRow) {
      const int row = L0 + n, col = o0 + t;
      if (domask && (row < 4 || row >= 508)) continue;
      const float er = Pf[row * 512 + col];
      const float ar = Pa[row * 512 + ao0 + t];
      Qf[row * 512 + col] = er * w + ar * (1.0f - w);
    } else {
      const int row = o0 + t, col = L0 + n;
      const float er = Pf[row * 512 + col];
      const float ar = Pa[(ao0 + t) * 512 + col];
      Qf[row * 512 + col] = er * w + ar * (1.0f - w);
    }
  }
#endif
}

extern "C" void kernel_launch(void* const* d_in, const int* in_sizes, int n_in,
                              void* d_out, int out_size, void* d_ws, size_t ws_size,
                              hipStream_t stream) {
  const float* x = (const float*)d_in[0];
  float* out = (float*)d_out;

  // out_size = 4*6*16*512*512 = 100,663,296 f32 -> 25,165,824 float4 chunks.
  const int n4 = out_size / 4;
  // 4096 blocks * 256 threads: each thread copies 24 float4 (grid-stride).
  bulk_copy_kernel<<<4096, 256, 0, stream>>>((const f4*)x, (f4*)out, n4);

  // 32768 waves (16 edges * 4 * 16 * 32 tiles) = 4096 blocks of 8 wave32s.
  // Runs after the copy on the same stream, overwriting the strips.
  edge_fixup_kernel<<<4096, 256, 0, stream>>>(x, out);
}